// VQC_31610959299041
// MI455X (gfx1250) — compile-verified
//
#include <hip/hip_runtime.h>
#include <math.h>

// VQC statevector simulator, fully fused, LDS-resident state.
// MI455X (gfx1250, wave32): 256 threads = 8 waves per block, 4 circuits/block.

#define NQ      10
#define NL      4
#define NC      4
#define NSTATES 1024          // 2^NQ
#define TILE_B  4             // circuits per workgroup (psi = 32 KB LDS)
#define BLOCK   256
#define NWAVES  (BLOCK / 32)  // wave32

typedef float v2f __attribute__((ext_vector_type(2)));
typedef float v8f __attribute__((ext_vector_type(8)));

__device__ __forceinline__ float2 cmul(float2 a, float2 b) {
  return make_float2(a.x * b.x - a.y * b.y, a.x * b.y + a.y * b.x);
}
__device__ __forceinline__ float2 cadd(float2 a, float2 b) {
  return make_float2(a.x + b.x, a.y + b.y);
}

__global__ __launch_bounds__(BLOCK)
void vqc_fused_kernel(const float* __restrict__ x,        // (B, NQ)
                      const float* __restrict__ weights,  // (NL, NQ, 3)
                      const float* __restrict__ bias,     // (NC)
                      const float* __restrict__ scale,    // (NC)
                      float* __restrict__ out)            // (B, NC)
{
  __shared__ float2 s_psi[TILE_B][NSTATES];        // 32 KB
  __shared__ float2 s_u[TILE_B][NQ][4];            // fused 1q gates U = Rot*RX
  __shared__ float  s_zpart[NWAVES][TILE_B][NC];   // per-wave WMMA partials

  const int tid = threadIdx.x;
  const int b0  = blockIdx.x * TILE_B;

  // ---- init |00..0> for each circuit in tile ----
  for (int g = tid; g < TILE_B * NSTATES; g += BLOCK) {
    int b = g >> 10, i = g & (NSTATES - 1);
    s_psi[b][i] = make_float2(i == 0 ? 1.0f : 0.0f, 0.0f);
  }

  for (int l = 0; l < NL; ++l) {
    __syncthreads();
    // ---- fused per-qubit gate U = Rot(phi,theta,omega) @ RX(x) ----
    if (tid < TILE_B * NQ) {
      int b = tid / NQ, q = tid % NQ;
      float hx = 0.5f * x[(b0 + b) * NQ + q];
      float cx = __cosf(hx), sx = __sinf(hx);
      const float* wr = weights + (l * NQ + q) * 3;
      float phi = wr[0], th = wr[1], om = wr[2];
      float ct  = __cosf(0.5f * th), st = __sinf(0.5f * th);
      float app = 0.5f * (phi + om), apm = 0.5f * (phi - om);
      float2 m00 = make_float2( __cosf(app) * ct, -__sinf(app) * ct);
      float2 m01 = make_float2(-__cosf(apm) * st, -__sinf(apm) * st);
      float2 m10 = make_float2( __cosf(apm) * st, -__sinf(apm) * st);
      float2 m11 = make_float2( __cosf(app) * ct,  __sinf(app) * ct);
      float2 x00 = make_float2(cx, 0.0f);          // RX diag
      float2 x01 = make_float2(0.0f, -sx);         // RX off-diag (== x10)
      s_u[b][q][0] = cadd(cmul(m00, x00), cmul(m01, x01));
      s_u[b][q][1] = cadd(cmul(m00, x01), cmul(m01, x00));
      s_u[b][q][2] = cadd(cmul(m10, x00), cmul(m11, x01));
      s_u[b][q][3] = cadd(cmul(m10, x01), cmul(m11, x00));
    }
    __syncthreads();

    // ---- apply 10 fused 1q gates (pairwise in-place updates in LDS) ----
    for (int q = 0; q < NQ; ++q) {
      for (int wk = tid; wk < TILE_B * (NSTATES / 2); wk += BLOCK) {
        int b  = wk >> 9, p = wk & 511;
        int i0 = ((p >> q) << (q + 1)) | (p & ((1 << q) - 1));
        int i1 = i0 | (1 << q);
        float2 a0 = s_psi[b][i0], a1 = s_psi[b][i1];
        float2 u00 = s_u[b][q][0], u01 = s_u[b][q][1];
        float2 u10 = s_u[b][q][2], u11 = s_u[b][q][3];
        s_psi[b][i0] = cadd(cmul(u00, a0), cmul(u01, a1));
        s_psi[b][i1] = cadd(cmul(u10, a0), cmul(u11, a1));
      }
      __syncthreads();
    }

    // ---- CNOT ring (0->1, .., 9->0) composed into ONE gather permutation:
    // new[i] = old[f0(f1(..f9(i)..))], f_q(j) = j ^ (bit_q(j) << ((q+1)%NQ))
    float2 stg[TILE_B * NSTATES / BLOCK];   // 16 complex in registers
    #pragma unroll
    for (int k = 0; k < TILE_B * NSTATES / BLOCK; ++k) {
      int g = tid + k * BLOCK;
      int b = g >> 10, i = g & (NSTATES - 1);
      int j = i;
      #pragma unroll
      for (int q = NQ - 1; q >= 0; --q) {
        int t = (q + 1) % NQ;
        j ^= ((j >> q) & 1) << t;
      }
      stg[k] = s_psi[b][j];
    }
    __syncthreads();
    #pragma unroll
    for (int k = 0; k < TILE_B * NSTATES / BLOCK; ++k) {
      int g = tid + k * BLOCK;
      s_psi[g >> 10][g & (NSTATES - 1)] = stg[k];
    }
  }
  __syncthreads();

  // ---- z[b][c] = sum_s |psi[b][s]|^2 * (1 - 2*bit_c(s))  via WMMA ----
  // A (16x4 f32): row M = lane&15 (batch, padded), K = vgpr + 2*(lane>>4)
  // B (4x16 f32): row K same mapping, col N = lane&15 (class, +-1, padded 0)
  // D (16x16):    vgpr r, lane-half h -> M = r + 8h, N = lane&15
  const int wave = tid >> 5;
  const int lane = tid & 31;
  const int lh   = lane >> 4;
  const int lm   = lane & 15;
  const int k0   = 2 * lh;
  const int   arow   = lm & (TILE_B - 1);                 // branchless A row
  const float amask  = (lm < TILE_B) ? 1.0f : 0.0f;       // zero padded rows
  const unsigned bvalid = (lm < NC) ? 0x3F800000u : 0u;   // +1.0f or +0.0f

  v8f acc = {0.f, 0.f, 0.f, 0.f, 0.f, 0.f, 0.f, 0.f};
  for (int step = 0; step < (NSTATES / NWAVES) / 4; ++step) {  // 32 WMMAs/wave
    int sb = wave * (NSTATES / NWAVES) + step * 4;
    v2f a, bmat;
    // A fragment: |psi|^2, unconditional LDS load (dup rows broadcast)
    float2 p0 = s_psi[arow][sb + k0 + 0];
    float2 p1 = s_psi[arow][sb + k0 + 1];
    a[0] = (p0.x * p0.x + p0.y * p0.y) * amask;
    a[1] = (p1.x * p1.x + p1.y * p1.y) * amask;
    // B fragment: sign = 1 - 2*bit_lm(s); padded lanes give (-)0.0f
    unsigned sg0 = ((unsigned)((sb + k0 + 0) >> lm) & 1u) << 31;
    unsigned sg1 = ((unsigned)((sb + k0 + 1) >> lm) & 1u) << 31;
    bmat[0] = __int_as_float((int)(bvalid | sg0));
    bmat[1] = __int_as_float((int)(bvalid | sg1));
    acc = __builtin_amdgcn_wmma_f32_16x16x4_f32(
        /*neg_a=*/false, a, /*neg_b=*/false, bmat,
        /*c_mod=*/(short)0, acc, /*reuse_a=*/false, /*reuse_b=*/false);
  }
  if (lh == 0 && lm < NC) {           // rows 0..3 live in vgprs 0..3, half 0
    #pragma unroll
    for (int r = 0; r < TILE_B; ++r)
      s_zpart[wave][r][lm] = acc[r];
  }
  __syncthreads();

  // ---- cross-wave reduce + bias/scale + softmax ----
  if (tid < TILE_B) {
    float z[NC], e[NC];
    float m = -1e30f;
    #pragma unroll
    for (int c = 0; c < NC; ++c) {
      float s = 0.f;
      #pragma unroll
      for (int w = 0; w < NWAVES; ++w) s += s_zpart[w][tid][c];
      z[c] = (s + bias[c]) * scale[c];
      m = fmaxf(m, z[c]);
    }
    float den = 0.f;
    #pragma unroll
    for (int c = 0; c < NC; ++c) { e[c] = __expf(z[c] - m); den += e[c]; }
    float inv = 1.0f / den;
    #pragma unroll
    for (int c = 0; c < NC; ++c)
      out[(b0 + tid) * NC + c] = e[c] * inv;
  }
}

extern "C" void kernel_launch(void* const* d_in, const int* in_sizes, int n_in,
                              void* d_out, int out_size, void* d_ws, size_t ws_size,
                              hipStream_t stream) {
  const float* x       = (const float*)d_in[0];
  const float* weights = (const float*)d_in[1];
  const float* bias    = (const float*)d_in[2];
  const float* scale   = (const float*)d_in[3];
  float* out           = (float*)d_out;

  const int Bsz = in_sizes[0] / NQ;   // 2048
  dim3 grid(Bsz / TILE_B);            // 512 workgroups
  dim3 block(BLOCK);                  // 8 wave32 waves
  vqc_fused_kernel<<<grid, block, 0, stream>>>(x, weights, bias, scale, out);
}